// OnChainGNN_77738908057668
// MI455X (gfx1250) — compile-verified
//
#include <hip/hip_runtime.h>
#include <hip/hip_bf16.h>

// ---------------------------------------------------------------------------
// OnChainGNN forward for MI455X (gfx1250, wave32).
//
// Input order (setup_inputs dict order, params dict flattened in insertion
// order):
//   0: nodes (N,8) f32        1: edge_attr (E,2) f32
//   2..22: per-layer params i=0..2: sage_w, sage_b, gat_wl, gat_wr, gat_a,
//          ln_s, ln_b   (7 per layer)
//   23: gat_we0 (2,512)  24: pool_w (256,1)  25: pool_b (1)
//   26: r1_w (768,512)   27: r1_b (512)  28: ln1_s  29: ln1_b
//   30: r2_w (512,256)   31: r2_b (256)  32: ln2_s  33: ln2_b
//   34: senders (E) i32  35: receivers (E) i32  36: graph_idx (N) i32
// Output: (G=4, 256) f32.
// ---------------------------------------------------------------------------

#define HEADS 4
#define GRAPHS 4
#define NEG_BIG (-3.402823466e38f)

typedef __attribute__((ext_vector_type(2))) float v2f;
typedef __attribute__((ext_vector_type(8))) float v8f;

// ---------------------------------------------------------------------------
// fp32 WMMA GEMM: Y(N,M) = X(N,K) @ W(K,M) + bias   (row-major everywhere)
// One wave32 per 16x64 output block: 4 accumulator tiles share a single A
// fragment per k-step (4x register reuse of A; 4 V_WMMA_F32_16X16X4_F32 per
// 4-wide k-step).  Fragment layouts per CDNA5 ISA:
//   A (16x4): lane l holds row M=l%16, K = vgpr + 2*(l>=16)
//   B (4x16): lane l holds col N=l%16, K = vgpr + 2*(l>=16)
//   C/D (16x16): vgpr j, lane l -> row = j + 8*(l>=16), col = l%16
// Requires: Nrows % 16 == 0, M % 64 == 0, K % 4 == 0 (true for all call
// sites).  Block-bound branch is wave-uniform, so EXEC stays all ones.
// ---------------------------------------------------------------------------
__global__ __launch_bounds__(256)
void gemm_wmma_f32(const float* __restrict__ X, const float* __restrict__ W,
                   const float* __restrict__ bias, float* __restrict__ Y,
                   int Nrows, int K, int M) {
  const int wave = threadIdx.x >> 5;     // 8 waves / block
  const int lane = threadIdx.x & 31;
  const int nblk = M >> 6;               // 64-wide column blocks
  const int total = (Nrows >> 4) * nblk;
  const int blk = blockIdx.x * 8 + wave;
  if (blk >= total) return;              // uniform across the wave
  const int tm = blk / nblk;
  const int tn = blk - tm * nblk;
  const int row0 = tm << 4, col0 = tn << 6;
  const int r  = lane & 15;              // row (A) / col (B) within tile
  const int hi = lane >> 4;              // K sub-phase (0 or 1)

  const float* __restrict__ arow = X + (long)(row0 + r) * K;
  const float* __restrict__ bcol = W + (col0 + r);

  v8f acc[4] = {};
  for (int k0 = 0; k0 < K; k0 += 4) {
    const int ka = k0 + (hi << 1);
    v2f a;
    a.x = arow[ka];
    a.y = arow[ka + 1];
    __builtin_prefetch(arow + ka + 16, 0, 1);   // global_prefetch_b8
    const float* b0 = bcol + (long)ka * M;
    const float* b1 = bcol + (long)(ka + 1) * M;
#pragma unroll
    for (int c = 0; c < 4; ++c) {
      v2f b;
      b.x = b0[16 * c];
      b.y = b1[16 * c];
      acc[c] = __builtin_amdgcn_wmma_f32_16x16x4_f32(
          /*neg_a=*/false, a, /*neg_b=*/false, b,
          /*c_mod=*/(short)0, acc[c], /*reuse_a=*/false, /*reuse_b=*/false);
    }
  }
#pragma unroll
  for (int c = 0; c < 4; ++c) {
    const int col = col0 + 16 * c + r;
    const float badd = bias ? bias[col] : 0.0f;
#pragma unroll
    for (int j = 0; j < 8; ++j)
      Y[(long)(row0 + j + 8 * hi) * M + col] = acc[c][j] + badd;
  }
}

// ---------------------------------------------------------------------------
__global__ void fill_f32(float* __restrict__ p, float v, int n) {
  int t = blockIdx.x * blockDim.x + threadIdx.x;
  if (t < n) p[t] = v;
}

__device__ __forceinline__ void atomicMaxF(float* addr, float val) {
  int* ia = reinterpret_cast<int*>(addr);
  int cur = __float_as_int(*addr);
  while (__int_as_float(cur) < val) {
    int assumed = cur;
    cur = atomicCAS(ia, assumed, __float_as_int(val));
    if (cur == assumed) break;
  }
}

__device__ __forceinline__ float eluf(float x) {
  return x > 0.0f ? x : (expf(x) - 1.0f);
}

// ------------------------- SAGE neighbor aggregation -----------------------
__global__ void sage_scatter(const float* __restrict__ x,
                             const int* __restrict__ snd,
                             const int* __restrict__ rcv,
                             float* __restrict__ nsum, float* __restrict__ cnt,
                             int E, int din) {
  int t = blockIdx.x * blockDim.x + threadIdx.x;
  if (t >= E * din) return;
  int e = t / din, j = t - e * din;
  int s = snd[e], r = rcv[e];
  atomicAdd(&nsum[(long)r * din + j], x[(long)s * din + j]);
  if (j == 0) atomicAdd(&cnt[r], 1.0f);
}

__global__ void sage_concat(const float* __restrict__ x,
                            const float* __restrict__ nsum,
                            const float* __restrict__ cnt,
                            float* __restrict__ xcat, int N, int din) {
  int t = blockIdx.x * blockDim.x + threadIdx.x;
  int w = 2 * din;
  if (t >= N * w) return;
  int n = t / w, j = t - n * w;
  float v;
  if (j < din) v = x[(long)n * din + j];
  else v = nsum[(long)n * din + (j - din)] / fmaxf(cnt[n], 1.0f);
  xcat[t] = v;
}

// ------------------------------- GATv2 edges -------------------------------
__global__ void gat_edge_logits(const float* __restrict__ xl,
                                const float* __restrict__ xr,
                                const float* __restrict__ att,   // (H,d)
                                const float* __restrict__ ea,    // (E,2) or null
                                const float* __restrict__ we,    // (2,H*d) or null
                                const int* __restrict__ snd,
                                const int* __restrict__ rcv,
                                float* __restrict__ logits,      // (E,H)
                                float* __restrict__ lmax,        // (N,H)
                                int E, int d) {
  int t = blockIdx.x * blockDim.x + threadIdx.x;
  if (t >= E * HEADS) return;
  int e = t / HEADS, h = t - e * HEADS;
  int s = snd[e], r = rcv[e];
  const float* xls = xl + (long)s * HEADS * d + h * d;
  const float* xrr = xr + (long)r * HEADS * d + h * d;
  float e0 = 0.f, e1 = 0.f;
  if (ea) { e0 = ea[2 * e]; e1 = ea[2 * e + 1]; }
  const float* a = att + h * d;
  float acc = 0.f;
  for (int j = 0; j < d; ++j) {
    float m = xls[j] + xrr[j];
    if (we) m += e0 * we[h * d + j] + e1 * we[HEADS * d + h * d + j];
    m = (m > 0.f) ? m : 0.2f * m;          // leaky_relu(0.2)
    acc = fmaf(m, a[j], acc);
  }
  logits[t] = acc;
  atomicMaxF(&lmax[r * HEADS + h], acc);
}

__global__ void gat_edge_exp(const int* __restrict__ rcv,
                             const float* __restrict__ lmax,
                             float* __restrict__ logits,   // in: logit, out: ex
                             float* __restrict__ ssum, int E) {
  int t = blockIdx.x * blockDim.x + threadIdx.x;
  if (t >= E * HEADS) return;
  int e = t / HEADS, h = t - e * HEADS;
  int r = rcv[e];
  float ex = expf(logits[t] - lmax[r * HEADS + h]);
  logits[t] = ex;
  atomicAdd(&ssum[r * HEADS + h], ex);
}

__global__ void gat_edge_agg(const float* __restrict__ xl,
                             const float* __restrict__ ex,
                             const float* __restrict__ ssum,
                             const int* __restrict__ snd,
                             const int* __restrict__ rcv,
                             float* __restrict__ agg, int E, int d) {
  int t = blockIdx.x * blockDim.x + threadIdx.x;
  if (t >= E * HEADS) return;
  int e = t / HEADS, h = t - e * HEADS;
  int s = snd[e], r = rcv[e];
  float w = ex[t] / (ssum[r * HEADS + h] + 1e-8f);
  const float* xls = xl + (long)s * HEADS * d + h * d;
  float* ag = agg + (long)r * HEADS * d + h * d;
  for (int j = 0; j < d; ++j) atomicAdd(&ag[j], xls[j] * w);
}

// head-mean over H, then LayerNorm + ELU.  One 256-thread block per node.
__global__ __launch_bounds__(256)
void head_mean_ln_elu(const float* __restrict__ agg,
                      const float* __restrict__ lns,
                      const float* __restrict__ lnb,
                      float* __restrict__ xout, int d) {
  const int n = blockIdx.x, j = threadIdx.x;
  __shared__ float red[256];
  __shared__ float stat[2];
  float v = 0.f;
  if (j < d) {
    const float* ag = agg + (long)n * HEADS * d;
    v = 0.25f * (ag[j] + ag[d + j] + ag[2 * d + j] + ag[3 * d + j]);
  }
  red[j] = (j < d) ? v : 0.f;
  __syncthreads();
  for (int off = 128; off; off >>= 1) { if (j < off) red[j] += red[j + off]; __syncthreads(); }
  if (j == 0) stat[0] = red[0] / (float)d;
  __syncthreads();
  float c = (j < d) ? (v - stat[0]) : 0.f;
  red[j] = c * c;
  __syncthreads();
  for (int off = 128; off; off >>= 1) { if (j < off) red[j] += red[j + off]; __syncthreads(); }
  if (j == 0) stat[1] = red[0] / (float)d;
  __syncthreads();
  if (j < d)
    xout[(long)n * d + j] = eluf(c * rsqrtf(stat[1] + 1e-6f) * lns[j] + lnb[j]);
}

// ------------------------------- pooling -----------------------------------
__global__ void pool_pass1(const float* __restrict__ x,
                           const int* __restrict__ gidx,
                           const float* __restrict__ pw, const float* __restrict__ pb,
                           float* __restrict__ msum, float* __restrict__ mmax,
                           float* __restrict__ gcnt, float* __restrict__ gate,
                           float* __restrict__ gmax, int N, int d) {
  int n = blockIdx.x * blockDim.x + threadIdx.x;
  if (n >= N) return;
  int g = gidx[n];
  const float* xn = x + (long)n * d;
  float acc = pb[0];
  for (int j = 0; j < d; ++j) {
    float v = xn[j];
    atomicAdd(&msum[g * d + j], v);
    atomicMaxF(&mmax[g * d + j], v);
    acc = fmaf(v, pw[j], acc);
  }
  gate[n] = acc;
  atomicAdd(&gcnt[g], 1.0f);
  atomicMaxF(&gmax[g], acc);
}

__global__ void pool_pass2(const int* __restrict__ gidx,
                           const float* __restrict__ gmax,
                           float* __restrict__ gate,      // in: gate, out: gexp
                           float* __restrict__ gsum, int N) {
  int n = blockIdx.x * blockDim.x + threadIdx.x;
  if (n >= N) return;
  int g = gidx[n];
  float ex = expf(gate[n] - gmax[g]);
  gate[n] = ex;
  atomicAdd(&gsum[g], ex);
}

__global__ void pool_pass3(const float* __restrict__ x,
                           const int* __restrict__ gidx,
                           const float* __restrict__ gexp,
                           const float* __restrict__ gsum,
                           float* __restrict__ attn, int N, int d) {
  int n = blockIdx.x * blockDim.x + threadIdx.x;
  if (n >= N) return;
  int g = gidx[n];
  float w = gexp[n] / (gsum[g] + 1e-8f);
  const float* xn = x + (long)n * d;
  for (int j = 0; j < d; ++j) atomicAdd(&attn[g * d + j], xn[j] * w);
}

__global__ void pool_finalize(const float* __restrict__ msum,
                              const float* __restrict__ mmax,
                              const float* __restrict__ attn,
                              const float* __restrict__ gcnt,
                              float* __restrict__ pooled, int d) {
  int t = blockIdx.x * blockDim.x + threadIdx.x;
  if (t >= GRAPHS * d) return;
  int g = t / d, j = t - g * d;
  float c = gcnt[g];
  pooled[g * 3 * d + j]         = msum[t] / fmaxf(c, 1.0f);
  pooled[g * 3 * d + d + j]     = (c > 0.f) ? mmax[t] : 0.f;
  pooled[g * 3 * d + 2 * d + j] = attn[t];
}

// ---------------------------- readout MLP (G=4) ----------------------------
__global__ __launch_bounds__(512)
void readout_mlp(const float* __restrict__ pooled,
                 const float* __restrict__ r1w, const float* __restrict__ r1b,
                 const float* __restrict__ ln1s, const float* __restrict__ ln1b,
                 const float* __restrict__ r2w, const float* __restrict__ r2b,
                 const float* __restrict__ ln2s, const float* __restrict__ ln2b,
                 float* __restrict__ out) {
  const int g = blockIdx.x, j = threadIdx.x;
  __shared__ float red[512];
  __shared__ float h1[512];
  __shared__ float stat[2];
  const float* p = pooled + g * 768;
  float acc = r1b[j];
  for (int k = 0; k < 768; ++k) acc = fmaf(p[k], r1w[k * 512 + j], acc);
  // LayerNorm(512) + ELU
  red[j] = acc; __syncthreads();
  for (int off = 256; off; off >>= 1) { if (j < off) red[j] += red[j + off]; __syncthreads(); }
  if (j == 0) stat[0] = red[0] * (1.0f / 512.0f);
  __syncthreads();
  float c = acc - stat[0];
  red[j] = c * c; __syncthreads();
  for (int off = 256; off; off >>= 1) { if (j < off) red[j] += red[j + off]; __syncthreads(); }
  if (j == 0) stat[1] = red[0] * (1.0f / 512.0f);
  __syncthreads();
  h1[j] = eluf(c * rsqrtf(stat[1] + 1e-6f) * ln1s[j] + ln1b[j]);
  __syncthreads();
  // second dense (512 -> 256) + LayerNorm(256)
  float acc2 = 0.f;
  if (j < 256) {
    acc2 = r2b[j];
    for (int k = 0; k < 512; ++k) acc2 = fmaf(h1[k], r2w[k * 256 + j], acc2);
  }
  red[j] = (j < 256) ? acc2 : 0.f; __syncthreads();
  for (int off = 256; off; off >>= 1) { if (j < off) red[j] += red[j + off]; __syncthreads(); }
  if (j == 0) stat[0] = red[0] * (1.0f / 256.0f);
  __syncthreads();
  float c2 = (j < 256) ? (acc2 - stat[0]) : 0.f;
  red[j] = c2 * c2; __syncthreads();
  for (int off = 256; off; off >>= 1) { if (j < off) red[j] += red[j + off]; __syncthreads(); }
  if (j == 0) stat[1] = red[0] * (1.0f / 256.0f);
  __syncthreads();
  if (j < 256)
    out[g * 256 + j] = c2 * rsqrtf(stat[1] + 1e-6f) * ln2s[j] + ln2b[j];
}

// ---------------------------------------------------------------------------
static inline void fill(float* p, float v, int n, hipStream_t s) {
  fill_f32<<<(n + 255) / 256, 256, 0, s>>>(p, v, n);
}

extern "C" void kernel_launch(void* const* d_in, const int* in_sizes, int n_in,
                              void* d_out, int out_size, void* d_ws, size_t ws_size,
                              hipStream_t stream) {
  const float* nodes     = (const float*)d_in[0];
  const float* edge_attr = (const float*)d_in[1];
  const float* P[32];
  for (int i = 0; i < 32; ++i) P[i] = (const float*)d_in[2 + i];
  const int* senders   = (const int*)d_in[34];
  const int* receivers = (const int*)d_in[35];
  const int* gidx      = (const int*)d_in[36];
  const int N = in_sizes[36];   // 20000
  const int E = in_sizes[34];   // 160000

  // workspace partition (floats)
  float* ws = (float*)d_ws;
  float* XL   = ws;                 // N*1024
  float* XR   = XL   + (long)N * 1024;  // N*1024 (reused as AGG)
  float* XCUR = XR   + (long)N * 1024;  // N*256
  float* SAGE = XCUR + (long)N * 256;   // N*256
  float* XCAT = SAGE + (long)N * 256;   // N*256
  float* NSUM = XCAT + (long)N * 256;   // N*128
  float* CNT  = NSUM + (long)N * 128;   // N
  float* EXB  = CNT  + (long)N;         // E*HEADS
  float* LMAX = EXB  + (long)E * HEADS; // N*HEADS
  float* SSUM = LMAX + (long)N * HEADS; // N*HEADS
  float* GATE = SSUM + (long)N * HEADS; // N
  float* MSUM = GATE + (long)N;         // G*256
  float* MMAX = MSUM + GRAPHS * 256;
  float* ATTN = MMAX + GRAPHS * 256;
  float* GCNT = ATTN + GRAPHS * 256;
  float* GMAX = GCNT + GRAPHS;
  float* GSUM = GMAX + GRAPHS;
  float* POOL = GSUM + GRAPHS;          // G*768

  struct Layer { int pbase, din, dout; };
  const Layer L[3] = { {0, 8, 128}, {7, 128, 128}, {14, 128, 256} };

  for (int i = 0; i < 3; ++i) {
    const float* sw  = P[L[i].pbase + 0];
    const float* sb  = P[L[i].pbase + 1];
    const float* wl  = P[L[i].pbase + 2];
    const float* wr  = P[L[i].pbase + 3];
    const float* att = P[L[i].pbase + 4];
    const float* lns = P[L[i].pbase + 5];
    const float* lnb = P[L[i].pbase + 6];
    const int din = L[i].din, dout = L[i].dout, HD = HEADS * dout;
    const float* xin = (i == 0) ? nodes : XCUR;

    // ---- GraphSAGE: nmean scatter + concat + WMMA GEMM ----
    fill(NSUM, 0.f, N * din, stream);
    fill(CNT, 0.f, N, stream);
    sage_scatter<<<(E * din + 255) / 256, 256, 0, stream>>>(
        xin, senders, receivers, NSUM, CNT, E, din);
    sage_concat<<<(N * 2 * din + 255) / 256, 256, 0, stream>>>(
        xin, NSUM, CNT, XCAT, N, din);
    {
      int blks = (N / 16) * (dout / 64);
      gemm_wmma_f32<<<(blks + 7) / 8, 256, 0, stream>>>(
          XCAT, sw, sb, SAGE, N, 2 * din, dout);
    }

    // ---- GATv2 projections (WMMA) ----
    {
      int blks = (N / 16) * (HD / 64);
      gemm_wmma_f32<<<(blks + 7) / 8, 256, 0, stream>>>(
          SAGE, wl, nullptr, XL, N, dout, HD);
      gemm_wmma_f32<<<(blks + 7) / 8, 256, 0, stream>>>(
          SAGE, wr, nullptr, XR, N, dout, HD);
    }

    // ---- edge softmax + aggregation ----
    fill(LMAX, NEG_BIG, N * HEADS, stream);
    gat_edge_logits<<<(E * HEADS + 255) / 256, 256, 0, stream>>>(
        XL, XR, att, (i == 0) ? edge_attr : nullptr, (i == 0) ? P[21] : nullptr,
        senders, receivers, EXB, LMAX, E, dout);
    fill(SSUM, 0.f, N * HEADS, stream);
    gat_edge_exp<<<(E * HEADS + 255) / 256, 256, 0, stream>>>(
        receivers, LMAX, EXB, SSUM, E);
    float* AGG = XR;                     // xr no longer needed; reuse as agg
    fill(AGG, 0.f, N * HD, stream);
    gat_edge_agg<<<(E * HEADS + 255) / 256, 256, 0, stream>>>(
        XL, EXB, SSUM, senders, receivers, AGG, E, dout);

    // ---- head mean + LayerNorm + ELU -> new node features ----
    head_mean_ln_elu<<<N, 256, 0, stream>>>(AGG, lns, lnb, XCUR, dout);
  }

  // ---- pooling (mean / max / attention) over 4 graphs ----
  fill(MSUM, 0.f, GRAPHS * 256, stream);
  fill(MMAX, NEG_BIG, GRAPHS * 256, stream);
  fill(ATTN, 0.f, GRAPHS * 256, stream);
  fill(GCNT, 0.f, GRAPHS, stream);
  fill(GMAX, NEG_BIG, GRAPHS, stream);
  fill(GSUM, 0.f, GRAPHS, stream);
  pool_pass1<<<(N + 255) / 256, 256, 0, stream>>>(
      XCUR, gidx, P[22], P[23], MSUM, MMAX, GCNT, GATE, GMAX, N, 256);
  pool_pass2<<<(N + 255) / 256, 256, 0, stream>>>(gidx, GMAX, GATE, GSUM, N);
  pool_pass3<<<(N + 255) / 256, 256, 0, stream>>>(XCUR, gidx, GATE, GSUM, ATTN, N, 256);
  pool_finalize<<<(GRAPHS * 256 + 255) / 256, 256, 0, stream>>>(
      MSUM, MMAX, ATTN, GCNT, POOL, 256);

  // ---- readout MLP ----
  readout_mlp<<<GRAPHS, 512, 0, stream>>>(
      POOL, P[24], P[25], P[26], P[27], P[28], P[29], P[30], P[31],
      (float*)d_out);
  (void)n_in; (void)out_size; (void)ws_size;
}